// SpatialOnlyEEGGCNN_75359496176060
// MI455X (gfx1250) — compile-verified
//
#include <hip/hip_runtime.h>
#include <hip/hip_bf16.h>

typedef float v2f __attribute__((ext_vector_type(2)));
typedef float v8f __attribute__((ext_vector_type(8)));

#define N_NODES  131072
#define N_EDGES  2097152
#define N_GRAPHS 2048
#define BN_EPS   1e-5f
#define SLOPE    0.01f

// ---------------------------------------------------------------------------
// Zero fill
// ---------------------------------------------------------------------------
__global__ void zero_kernel(float* __restrict__ p, long n) {
    long i = (long)blockIdx.x * blockDim.x + threadIdx.x;
    if (i < n) p[i] = 0.0f;
}

// ---------------------------------------------------------------------------
// Degree over dst (self loop handled as +1 in rsqrt kernel)
// ---------------------------------------------------------------------------
__global__ void degree_kernel(const int* __restrict__ ei, float* __restrict__ deg) {
    int e = blockIdx.x * blockDim.x + threadIdx.x;
    if (e < N_EDGES) {
        int dst = ei[N_EDGES + e];
        atomicAdd(&deg[dst], 1.0f);
    }
}

__global__ void dinv_kernel(float* __restrict__ deg) {
    int i = blockIdx.x * blockDim.x + threadIdx.x;
    if (i < N_NODES) deg[i] = rsqrtf(deg[i] + 1.0f);  // +1 self loop
}

// ---------------------------------------------------------------------------
// FP32 WMMA GEMM:  C[nrows x NC] = A[nrows x K] @ B[K x NC]
// One wave computes a 16-row stripe (NC/16 tiles of 16x16), K stepped by 4
// via V_WMMA_F32_16X16X4_F32. B staged in LDS with async b128 copies
// (GLOBAL_LOAD_ASYNC_TO_LDS_B128, ASYNCcnt). 128 threads = 4 waves/block.
// ---------------------------------------------------------------------------
template <int K, int NC>
__global__ __launch_bounds__(128) void gemm_wmma_kernel(
    const float* __restrict__ A, const float* __restrict__ Bm,
    float* __restrict__ Cm) {
    constexpr int NT = NC / 16;
    __shared__ float Bs[K * NC];

    // async stage B (K*NC floats) into LDS, 16B per lane per op
    constexpr int NV4 = (K * NC) / 4;
    for (int i = threadIdx.x; i < NV4; i += 128) {
        unsigned lds_off = (unsigned)(uintptr_t)(&Bs[i * 4]);
        const float* gsrc = Bm + (long)i * 4;
        asm volatile("global_load_async_to_lds_b128 %0, %1, off"
                     :: "v"(lds_off), "v"(gsrc) : "memory");
    }
    asm volatile("s_wait_asynccnt 0" ::: "memory");
    __syncthreads();

    const int wave = threadIdx.x >> 5;
    const int lane = threadIdx.x & 31;
    const int half = lane >> 4;     // 0: K=0,1 ; 1: K=2,3 (per 32-bit 16x4 layout)
    const int r    = lane & 15;

    const long row0 = ((long)blockIdx.x * 4 + wave) * 16;
    const float* __restrict__ Arow = A + row0 * K;

    v8f acc[NT] = {};
    for (int kk = 0; kk < K; kk += 4) {
        v2f a;
        a.x = Arow[(long)r * K + kk + 2 * half + 0];
        a.y = Arow[(long)r * K + kk + 2 * half + 1];
#pragma unroll
        for (int t = 0; t < NT; ++t) {
            v2f b;
            b.x = Bs[(kk + 2 * half + 0) * NC + t * 16 + r];
            b.y = Bs[(kk + 2 * half + 1) * NC + t * 16 + r];
            acc[t] = __builtin_amdgcn_wmma_f32_16x16x4_f32(
                false, a, false, b, (short)0, acc[t], false, false);
        }
    }
#pragma unroll
    for (int t = 0; t < NT; ++t) {
#pragma unroll
        for (int v = 0; v < 8; ++v) {
            Cm[(row0 + v + 8 * half) * NC + t * 16 + r] = acc[t][v];
        }
    }
}

// ---------------------------------------------------------------------------
// Edge aggregation: agg[dst] += xw[src] * dinv[src]*dinv[dst]
// float4 per lane (b128 gather); C/4 lanes per edge; f32 atomics for scatter.
// ---------------------------------------------------------------------------
template <int C>
__global__ __launch_bounds__(256) void edge_aggregate_kernel(
    const float* __restrict__ xw, const int* __restrict__ ei,
    const float* __restrict__ dinv, float* __restrict__ agg) {
    constexpr int TPE = C / 4;       // threads per edge
    constexpr int EPB = 256 / TPE;   // edges per block
    int e = blockIdx.x * EPB + threadIdx.x / TPE;
    int f = (threadIdx.x % TPE) * 4;
    if (e >= N_EDGES) return;
    int src = ei[e];
    int dst = ei[N_EDGES + e];
    float nrm = dinv[src] * dinv[dst];
    float4 v = *(const float4*)(xw + (long)src * C + f);
    float* ap = agg + (long)dst * C + f;
    atomicAdd(ap + 0, v.x * nrm);
    atomicAdd(ap + 1, v.y * nrm);
    atomicAdd(ap + 2, v.z * nrm);
    atomicAdd(ap + 3, v.w * nrm);
}

// ---------------------------------------------------------------------------
// Self loop + bias (vectorized):  agg[i] += xw[i] * dinv[n]^2 + b[f]
// ---------------------------------------------------------------------------
template <int C>
__global__ void finish_conv_kernel(float* __restrict__ agg,
                                   const float* __restrict__ xw,
                                   const float* __restrict__ dinv,
                                   const float* __restrict__ bias) {
    long i4 = (long)blockIdx.x * blockDim.x + threadIdx.x;
    if (i4 >= (long)N_NODES * C / 4) return;
    long base = i4 * 4;
    int n = (int)(base / C);
    int f = (int)(base % C);
    float di = dinv[n];
    float s = di * di;
    float4 a = ((const float4*)agg)[i4];
    float4 w = ((const float4*)xw)[i4];
    float4 b = *(const float4*)(bias + f);
    a.x += w.x * s + b.x;
    a.y += w.y * s + b.y;
    a.z += w.z * s + b.z;
    a.w += w.w * s + b.w;
    ((float4*)agg)[i4] = a;
}

// ---------------------------------------------------------------------------
// BN stats: per-channel sum / sumsq, block-local LDS reduce then atomics
// ---------------------------------------------------------------------------
template <int C>
__global__ __launch_bounds__(256) void bn_stats_kernel(
    const float* __restrict__ h, float* __restrict__ sum,
    float* __restrict__ sumsq) {
    constexpr int R = 256 / C;
    __shared__ float ss[256];
    __shared__ float sq[256];
    const int f = threadIdx.x % C;
    const int g = threadIdx.x / C;
    float s = 0.f, q = 0.f;
    for (int n = blockIdx.x * R + g; n < N_NODES; n += gridDim.x * R) {
        float v = h[(long)n * C + f];
        s += v;
        q += v * v;
    }
    ss[threadIdx.x] = s;
    sq[threadIdx.x] = q;
    __syncthreads();
    if (threadIdx.x < C) {
        float as = ss[threadIdx.x];
        float aq = sq[threadIdx.x];
#pragma unroll
        for (int i = 1; i < R; ++i) {
            as += ss[threadIdx.x + i * C];
            aq += sq[threadIdx.x + i * C];
        }
        atomicAdd(&sum[f], as);
        atomicAdd(&sumsq[f], aq);
    }
}

// ---------------------------------------------------------------------------
// BN apply (+ leaky ReLU), vectorized, in place
// ---------------------------------------------------------------------------
template <int C>
__global__ void bn_apply_kernel(float* __restrict__ h,
                                const float* __restrict__ sum,
                                const float* __restrict__ sumsq,
                                const float* __restrict__ gamma,
                                const float* __restrict__ beta) {
    long i4 = (long)blockIdx.x * blockDim.x + threadIdx.x;
    if (i4 >= (long)N_NODES * C / 4) return;
    int f = (int)((i4 * 4) % C);
    const float inv_n = 1.0f / (float)N_NODES;
    float4 sm = *(const float4*)(sum + f);
    float4 qm = *(const float4*)(sumsq + f);
    float4 gm = *(const float4*)(gamma + f);
    float4 bt = *(const float4*)(beta + f);
    float4 v = ((const float4*)h)[i4];
    float mu, var, y;
    mu = sm.x * inv_n; var = qm.x * inv_n - mu * mu;
    y = gm.x * (v.x - mu) * rsqrtf(var + BN_EPS) + bt.x; v.x = (y > 0.f) ? y : SLOPE * y;
    mu = sm.y * inv_n; var = qm.y * inv_n - mu * mu;
    y = gm.y * (v.y - mu) * rsqrtf(var + BN_EPS) + bt.y; v.y = (y > 0.f) ? y : SLOPE * y;
    mu = sm.z * inv_n; var = qm.z * inv_n - mu * mu;
    y = gm.z * (v.z - mu) * rsqrtf(var + BN_EPS) + bt.z; v.z = (y > 0.f) ? y : SLOPE * y;
    mu = sm.w * inv_n; var = qm.w * inv_n - mu * mu;
    y = gm.w * (v.w - mu) * rsqrtf(var + BN_EPS) + bt.w; v.w = (y > 0.f) ? y : SLOPE * y;
    ((float4*)h)[i4] = v;
}

// ---------------------------------------------------------------------------
// Pool: sum per graph + counts (batch is node->graph map), b128 gather
// ---------------------------------------------------------------------------
__global__ __launch_bounds__(256) void pool_kernel(
    const float* __restrict__ h2, const int* __restrict__ batch,
    float* __restrict__ pooled, float* __restrict__ cnt) {
    long i4 = (long)blockIdx.x * 256 + threadIdx.x;  // N_NODES*32 total
    int n = (int)(i4 >> 5);
    int f = (int)(i4 & 31) * 4;
    int g = batch[n];
    float4 v = ((const float4*)h2)[i4];
    float* pp = pooled + (long)g * 128 + f;
    atomicAdd(pp + 0, v.x);
    atomicAdd(pp + 1, v.y);
    atomicAdd(pp + 2, v.z);
    atomicAdd(pp + 3, v.w);
    if (f == 0) atomicAdd(&cnt[g], 1.0f);
}

// ---------------------------------------------------------------------------
// Final: out[g] = (pooled[g]/max(cnt,1)) . Wo + bo
// ---------------------------------------------------------------------------
__global__ __launch_bounds__(128) void final_kernel(
    const float* __restrict__ pooled, const float* __restrict__ cnt,
    const float* __restrict__ Wo, const float* __restrict__ bo,
    float* __restrict__ out) {
    __shared__ float red[128];
    int g = blockIdx.x;
    int t = threadIdx.x;
    float c = cnt[g];
    c = (c < 1.f) ? 1.f : c;
    red[t] = pooled[(long)g * 128 + t] / c * Wo[t];
    __syncthreads();
    for (int s = 64; s > 0; s >>= 1) {
        if (t < s) red[t] += red[t + s];
        __syncthreads();
    }
    if (t == 0) out[g] = red[0] + bo[0];
}

// ---------------------------------------------------------------------------
// Launch
// ---------------------------------------------------------------------------
extern "C" void kernel_launch(void* const* d_in, const int* in_sizes, int n_in,
                              void* d_out, int out_size, void* d_ws, size_t ws_size,
                              hipStream_t stream) {
    const float* x      = (const float*)d_in[0];   // [131072,128]
    const int*   ei     = (const int*)  d_in[1];   // [2,2097152]
    const int*   batch  = (const int*)  d_in[2];   // [131072]
    const float* W1     = (const float*)d_in[3];   // [128,64]
    const float* b1     = (const float*)d_in[4];
    const float* gamma1 = (const float*)d_in[5];
    const float* beta1  = (const float*)d_in[6];
    const float* W2     = (const float*)d_in[7];   // [64,128]
    const float* b2     = (const float*)d_in[8];
    const float* gamma2 = (const float*)d_in[9];
    const float* beta2  = (const float*)d_in[10];
    const float* Wo     = (const float*)d_in[11];  // [128,1]
    const float* bo     = (const float*)d_in[12];
    float* out = (float*)d_out;                    // [2048,1]

    // workspace layout (floats)
    float* ws = (float*)d_ws;
    size_t off = 0;
    float* dinv  = ws + off; off += (size_t)N_NODES;          // deg -> d^-1/2
    float* xw1   = ws + off; off += (size_t)N_NODES * 64;
    float* h1    = ws + off; off += (size_t)N_NODES * 64;     // agg1 -> h1
    float* xw2   = ws + off; off += (size_t)N_NODES * 128;
    float* h2    = ws + off; off += (size_t)N_NODES * 128;    // agg2 -> h2
    float* s1    = ws + off; off += 64;
    float* q1    = ws + off; off += 64;
    float* s2    = ws + off; off += 128;
    float* q2    = ws + off; off += 128;
    float* pooled= ws + off; off += (size_t)N_GRAPHS * 128;
    float* cnt   = ws + off; off += (size_t)N_GRAPHS;

    auto zero = [&](float* p, long n) {
        zero_kernel<<<(int)((n + 255) / 256), 256, 0, stream>>>(p, n);
    };

    // ---- zero accumulators (harness does not re-poison between replays) ----
    zero(dinv, N_NODES);
    zero(h1,  (long)N_NODES * 64);
    zero(h2,  (long)N_NODES * 128);
    zero(s1,  64 + 64 + 128 + 128 + (long)N_GRAPHS * 128 + N_GRAPHS); // stats..cnt contiguous

    // ---- normalization coefficients ----
    degree_kernel<<<N_EDGES / 256, 256, 0, stream>>>(ei, dinv);
    dinv_kernel<<<N_NODES / 256, 256, 0, stream>>>(dinv);

    // ---- layer 1: GEMM (WMMA f32) -> edge aggregation -> self+bias -> BN -> lrelu
    gemm_wmma_kernel<128, 64><<<N_NODES / 64, 128, 0, stream>>>(x, W1, xw1);
    edge_aggregate_kernel<64><<<N_EDGES / 16, 256, 0, stream>>>(xw1, ei, dinv, h1);
    finish_conv_kernel<64><<<(N_NODES * 64 / 4) / 256, 256, 0, stream>>>(h1, xw1, dinv, b1);
    bn_stats_kernel<64><<<512, 256, 0, stream>>>(h1, s1, q1);
    bn_apply_kernel<64><<<(N_NODES * 64 / 4) / 256, 256, 0, stream>>>(h1, s1, q1, gamma1, beta1);

    // ---- layer 2 ----
    gemm_wmma_kernel<64, 128><<<N_NODES / 64, 128, 0, stream>>>(h1, W2, xw2);
    edge_aggregate_kernel<128><<<N_EDGES / 8, 256, 0, stream>>>(xw2, ei, dinv, h2);
    finish_conv_kernel<128><<<(N_NODES * 128 / 4) / 256, 256, 0, stream>>>(h2, xw2, dinv, b2);
    bn_stats_kernel<128><<<512, 256, 0, stream>>>(h2, s2, q2);
    bn_apply_kernel<128><<<(N_NODES * 128 / 4) / 256, 256, 0, stream>>>(h2, s2, q2, gamma2, beta2);

    // ---- pooling + readout ----
    pool_kernel<<<(N_NODES * 32) / 256, 256, 0, stream>>>(h2, batch, pooled, cnt);
    final_kernel<<<N_GRAPHS, 128, 0, stream>>>(pooled, cnt, Wo, bo, out);

    (void)in_sizes; (void)n_in; (void)out_size; (void)ws_size;
}